// DeepLK_36275293782297
// MI455X (gfx1250) — compile-verified
//
#include <hip/hip_runtime.h>
#include <math.h>

// DeepLK homography LK solver for MI455X (gfx1250, wave32).
// Hessian J*J^T reduction runs on v_wmma_f32_16x16x4_f32 using a
// layout-robust scheme: A rows = 16 (i,j) Hessian pairs, K = 4 pixels,
// B = all-ones (invariant under any B register layout), C read per the
// documented 16x16 f32 C layout. Warp/residual pass is gather-bound VALU.

#define BATCH 16
#define CHN   3
#define IW    384
#define IH    384
#define HW    (IW * IH)       // 147456
#define NEL   (CHN * HW)      // 442368
#define CENTER 191.5f
#define MAXIT 10

typedef __attribute__((ext_vector_type(2))) float v2f;
typedef __attribute__((ext_vector_type(8))) float v8f;

// workspace layout (floats)
#define WS_HESS 0            // [16][64]
#define WS_INVH 1024         // [16][64]
#define WS_P    2048         // [16][8]
#define WS_DP   2176         // [16][8]
#define WS_G    2304         // [MAXIT][16][8]
#define WS_TOTAL (WS_G + MAXIT * BATCH * 8)   // 3584 floats = 14 KiB

// ---------------------------------------------------------------- init
__global__ void dlk_init(float* __restrict__ ws) {
  for (int i = blockIdx.x * blockDim.x + threadIdx.x; i < WS_TOTAL;
       i += gridDim.x * blockDim.x) {
    ws[i] = (i >= WS_DP && i < WS_DP + BATCH * 8) ? 1.0f : 0.0f;  // dp0 = 1
  }
}

// central-difference gradient with edge replication
__device__ __forceinline__ void dlk_grad(const float* __restrict__ Tc, int x,
                                         int y, float& gx, float& gy) {
  int xm = (x > 0) ? x - 1 : 0;
  int xp = (x < IW - 1) ? x + 1 : IW - 1;
  int ym = (y > 0) ? y - 1 : 0;
  int yp = (y < IH - 1) ? y + 1 : IH - 1;
  unsigned row = (unsigned)(y * IW);
  gx = 0.5f * (Tc[row + xp] - Tc[row + xm]);
  gy = 0.5f * (Tc[(unsigned)(yp * IW + x)] - Tc[(unsigned)(ym * IW + x)]);
}

// ------------------------------------------------------- Hessian (WMMA)
// Hess[b][i][j] = sum over elements e of J_i(e)*J_j(e).
// Per wave iteration: 4 elements (K=0..3). WMMA w (w=0..3) has
// A[m,k] = J_i(e_k)*J_j(e_k) with pair p = w*16+m, i=p>>3, j=p&7.
// B = ones -> C[m,n] = K-sum, identical across n; read column N=0.
// A layout (documented): lane l holds M=l%16, VGPR jj holds K=jj+2*(l>>4).
__global__ void dlk_hessian(const float* __restrict__ temp,
                            float* __restrict__ hess) {
  const int b = blockIdx.y;
  const float* T0 = temp + b * NEL;
  const int lane = threadIdx.x & 31;
  const int wavesPerBatch = gridDim.x * (blockDim.x >> 5);
  // scalarize so the WMMA loop is uniform (EXEC stays all-1s)
  const int waveId = __builtin_amdgcn_readfirstlane(
      blockIdx.x * (blockDim.x >> 5) + (threadIdx.x >> 5));
  const int m = lane & 15;             // A-matrix row for this lane
  const int kset = lane >> 4;          // 0: K=0,1  1: K=2,3
  const int jcomp = m & 7;             // j component (same for all 4 pairs)
  const bool b0 = (jcomp & 1) != 0;    // select-tree masks, hoisted
  const bool b1 = (jcomp & 2) != 0;
  const bool b2 = (jcomp & 4) != 0;
  const bool ib = ((m >> 3) & 1) != 0; // i = (m>>3) + 2w

  v8f acc0 = {0.f, 0.f, 0.f, 0.f, 0.f, 0.f, 0.f, 0.f};
  v8f acc1 = acc0, acc2 = acc0, acc3 = acc0;
  const v2f ones = {1.0f, 1.0f};
  const int ngroups = NEL / 4;         // 110592

  for (int gidx = waveId; gidx < ngroups; gidx += wavesPerBatch) {
    v2f a0, a1, a2, a3;
#pragma unroll
    for (int jj = 0; jj < 2; ++jj) {
      int e = gidx * 4 + jj + 2 * kset;   // element (channel-major)
      int c = e / HW;
      int pix = e - c * HW;
      int y = pix / IW;
      int x = pix - y * IW;
      const float* Tc = T0 + c * HW;
      float X = (float)x - CENTER, Y = (float)y - CENTER;
      float gx, gy;
      dlk_grad(Tc, x, y, gx, gy);
      float j0 = X * gx, j1 = Y * gx, j2 = gx;
      float j3 = X * gy, j4 = Y * gy, j5 = gy;
      float j6 = -(X * j0 + Y * j3);
      float j7 = -(X * j1 + Y * j4);
      // branch-free select of J_jcomp (v_cndmask tree)
      float t0 = b0 ? j1 : j0;
      float t1 = b0 ? j3 : j2;
      float t2 = b0 ? j5 : j4;
      float t3 = b0 ? j7 : j6;
      float u0 = b1 ? t1 : t0;
      float u1 = b1 ? t3 : t2;
      float Jj = b2 ? u1 : u0;
      // i components: (m>>3)+2w -> pairwise single cndmask each
      float Ji0 = ib ? j1 : j0;
      float Ji1 = ib ? j3 : j2;
      float Ji2 = ib ? j5 : j4;
      float Ji3 = ib ? j7 : j6;
      a0[jj] = Ji0 * Jj;
      a1[jj] = Ji1 * Jj;
      a2[jj] = Ji2 * Jj;
      a3[jj] = Ji3 * Jj;
    }
    acc0 = __builtin_amdgcn_wmma_f32_16x16x4_f32(false, a0, false, ones,
                                                 (short)0, acc0, false, false);
    acc1 = __builtin_amdgcn_wmma_f32_16x16x4_f32(false, a1, false, ones,
                                                 (short)0, acc1, false, false);
    acc2 = __builtin_amdgcn_wmma_f32_16x16x4_f32(false, a2, false, ones,
                                                 (short)0, acc2, false, false);
    acc3 = __builtin_amdgcn_wmma_f32_16x16x4_f32(false, a3, false, ones,
                                                 (short)0, acc3, false, false);
  }

  // C layout: vgpr v, lanes 0-15 -> (M=v, N=lane); lanes 16-31 -> (M=v+8).
  // Column N=0 lives in lane 0 (M=v) and lane 16 (M=v+8).
  __shared__ float sh[64];
  for (int i = threadIdx.x; i < 64; i += blockDim.x) sh[i] = 0.0f;
  __syncthreads();
  if (lane == 0 || lane == 16) {
    int base = (lane == 16) ? 8 : 0;
#pragma unroll
    for (int v = 0; v < 8; ++v) {
      atomicAdd(&sh[0 * 16 + base + v], acc0[v]);
      atomicAdd(&sh[1 * 16 + base + v], acc1[v]);
      atomicAdd(&sh[2 * 16 + base + v], acc2[v]);
      atomicAdd(&sh[3 * 16 + base + v], acc3[v]);
    }
  }
  __syncthreads();
  for (int i = threadIdx.x; i < 64; i += blockDim.x)
    unsafeAtomicAdd(&hess[b * 64 + i], sh[i]);
}

// ------------------------------------------------- 8x8 inverse (per batch)
__global__ void dlk_invert(const float* __restrict__ hess,
                           float* __restrict__ invH) {
  int b = threadIdx.x;
  if (b >= BATCH) return;
  float M[8][16];
  for (int i = 0; i < 8; ++i)
    for (int j = 0; j < 8; ++j) {
      M[i][j] = hess[b * 64 + i * 8 + j];
      M[i][8 + j] = (i == j) ? 1.0f : 0.0f;
    }
  for (int i = 0; i < 8; ++i) {
    float inv = 1.0f / M[i][i];           // SPD Hessian: diagonal pivot
    for (int j = 0; j < 16; ++j) M[i][j] *= inv;
    for (int r = 0; r < 8; ++r) {
      if (r == i) continue;
      float f = M[r][i];
      for (int j = 0; j < 16; ++j) M[r][j] -= f * M[i][j];
    }
  }
  for (int i = 0; i < 8; ++i)
    for (int j = 0; j < 8; ++j) invH[b * 64 + i * 8 + j] = M[i][8 + j];
}

// -------------------------------- warp + residual + g = dIdp^T r (per iter)
__global__ void dlk_warp_accum(const float* __restrict__ img,
                               const float* __restrict__ temp,
                               const float* __restrict__ p,
                               float* __restrict__ g,
                               const int* __restrict__ max_itr, int it) {
  if (it >= *max_itr) return;
  const int b = blockIdx.y;
  const float* P = p + b * 8;
  const float h00 = P[0] + 1.0f, h01 = P[1], h02 = P[2];
  const float h10 = P[3], h11 = P[4] + 1.0f, h12 = P[5];
  const float h20 = P[6], h21 = P[7];
  const float* I0 = img + b * NEL;
  const float* T0 = temp + b * NEL;
  const float lim = 1.0f - 2.0f / (float)IW;   // W == H

  float a0 = 0, a1 = 0, a2 = 0, a3 = 0, a4 = 0, a5 = 0, a6 = 0, a7 = 0;

  for (int pix = blockIdx.x * blockDim.x + threadIdx.x; pix < HW;
       pix += gridDim.x * blockDim.x) {
    int y = pix / IW;
    int x = pix - y * IW;
    float X = (float)x - CENTER, Y = (float)y - CENTER;
    float zw = h20 * X + h21 * Y + 1.0f;
    float iz = 1.0f / zw;
    float Xw = (h00 * X + h01 * Y + h02) * iz + CENTER;
    float Yw = (h10 * X + h11 * Y + h12) * iz + CENTER;

    float x0f = floorf(Xw), y0f = floorf(Yw);
    float wx1 = Xw - x0f, wy1 = Yw - y0f;
    float wx0 = 1.0f - wx1, wy0 = 1.0f - wy1;
    int ix0 = (int)x0f, iy0 = (int)y0f;
    int ix1 = ix0 + 1, iy1 = iy0 + 1;
    bool vx0 = (ix0 >= 0) & (ix0 <= IW - 1);
    bool vx1 = (ix1 >= 0) & (ix1 <= IW - 1);
    bool vy0 = (iy0 >= 0) & (iy0 <= IH - 1);
    bool vy1 = (iy1 >= 0) & (iy1 <= IH - 1);
    int cx0 = min(max(ix0, 0), IW - 1), cx1 = min(max(ix1, 0), IW - 1);
    int cy0 = min(max(iy0, 0), IH - 1), cy1 = min(max(iy1, 0), IH - 1);
    float w00 = wy0 * wx0 * ((vx0 & vy0) ? 1.0f : 0.0f);
    float w01 = wy0 * wx1 * ((vx1 & vy0) ? 1.0f : 0.0f);
    float w10 = wy1 * wx0 * ((vx0 & vy1) ? 1.0f : 0.0f);
    float w11 = wy1 * wx1 * ((vx1 & vy1) ? 1.0f : 0.0f);
    unsigned o00 = (unsigned)(cy0 * IW + cx0), o01 = (unsigned)(cy0 * IW + cx1);
    unsigned o10 = (unsigned)(cy1 * IW + cx0), o11 = (unsigned)(cy1 * IW + cx1);

    float xn = Xw / CENTER - 1.0f, yn = Yw / CENTER - 1.0f;
    float mask = ((xn > -lim) & (xn < lim) & (yn > -lim) & (yn < lim))
                     ? 1.0f : 0.0f;
#pragma unroll
    for (int c = 0; c < CHN; ++c) {
      const float* Ic = I0 + c * HW;
      const float* Tc = T0 + c * HW;
      float Fw = w00 * Ic[o00] + w01 * Ic[o01] + w10 * Ic[o10] + w11 * Ic[o11];
      float r = Fw - Tc[(unsigned)pix] * mask;
      float gx, gy;
      dlk_grad(Tc, x, y, gx, gy);
      float j0 = X * gx, j1 = Y * gx, j3 = X * gy, j4 = Y * gy;
      a0 += j0 * r;  a1 += j1 * r;  a2 += gx * r;
      a3 += j3 * r;  a4 += j4 * r;  a5 += gy * r;
      a6 += -(X * j0 + Y * j3) * r;
      a7 += -(X * j1 + Y * j4) * r;
    }
  }

  // wave32 shuffle reduce, then cross-wave LDS, then global f32 atomics
  float acc[8] = {a0, a1, a2, a3, a4, a5, a6, a7};
#pragma unroll
  for (int j = 0; j < 8; ++j)
    for (int off = 16; off > 0; off >>= 1)
      acc[j] += __shfl_down(acc[j], off, 32);

  __shared__ float sh[8];
  if (threadIdx.x < 8) sh[threadIdx.x] = 0.0f;
  __syncthreads();
  if ((threadIdx.x & 31) == 0) {
#pragma unroll
    for (int j = 0; j < 8; ++j) atomicAdd(&sh[j], acc[j]);
  }
  __syncthreads();
  if (threadIdx.x < 8)
    unsafeAtomicAdd(&g[b * 8 + threadIdx.x], sh[threadIdx.x]);
}

// ---------------------------------------- dp = gate * (invH g); p -= dp
__global__ void dlk_solve(const float* __restrict__ invH,
                          const float* __restrict__ g, float* __restrict__ p,
                          float* __restrict__ dp,
                          const int* __restrict__ max_itr, int it) {
  int b = threadIdx.x;
  if (b >= BATCH) return;
  if (it >= *max_itr) return;
  const float* iH = invH + b * 64;
  const float* gb = g + b * 8;
  float* pb = p + b * 8;
  float* dpb = dp + b * 8;
  float n2 = 0.0f;
  for (int j = 0; j < 8; ++j) n2 += dpb[j] * dpb[j];
  float gate = (sqrtf(n2) > 1e-3f) ? 1.0f : 0.0f;
  float nd[8];
  for (int i = 0; i < 8; ++i) {
    float s = 0.0f;
    for (int j = 0; j < 8; ++j) s += iH[i * 8 + j] * gb[j];
    nd[i] = s;
  }
  nd[6] = 0.0f;
  nd[7] = 0.0f;
  for (int j = 0; j < 8; ++j) {
    float d = gate * nd[j];
    dpb[j] = d;
    pb[j] -= d;
  }
}

// ------------------------------------------------- out = (p[16,8], H[16,3,3])
__global__ void dlk_finalize(const float* __restrict__ p,
                             float* __restrict__ out) {
  int t = threadIdx.x;
  if (t < BATCH * 8) {
    out[t] = p[t];
  } else if (t < BATCH * 8 + BATCH * 9) {
    int idx = t - BATCH * 8;
    int b = idx / 9, ij = idx - b * 9;
    float v = (ij < 8) ? p[b * 8 + ij] : 0.0f;
    int i = ij / 3, j = ij - i * 3;
    out[t] = v + ((i == j) ? 1.0f : 0.0f);
  }
}

extern "C" void kernel_launch(void* const* d_in, const int* in_sizes, int n_in,
                              void* d_out, int out_size, void* d_ws,
                              size_t ws_size, hipStream_t stream) {
  const float* img = (const float*)d_in[0];
  const float* temp = (const float*)d_in[1];
  const int* max_itr = (const int*)d_in[2];
  float* ws = (float*)d_ws;
  float* hess = ws + WS_HESS;
  float* invH = ws + WS_INVH;
  float* p = ws + WS_P;
  float* dp = ws + WS_DP;
  float* gbuf = ws + WS_G;
  float* out = (float*)d_out;

  dlk_init<<<14, 256, 0, stream>>>(ws);
  dlk_hessian<<<dim3(16, BATCH), 256, 0, stream>>>(temp, hess);
  dlk_invert<<<1, 32, 0, stream>>>(hess, invH);
  for (int it = 0; it < MAXIT; ++it) {
    dlk_warp_accum<<<dim3(128, BATCH), 256, 0, stream>>>(
        img, temp, p, gbuf + it * BATCH * 8, max_itr, it);
    dlk_solve<<<1, 32, 0, stream>>>(invH, gbuf + it * BATCH * 8, p, dp,
                                    max_itr, it);
  }
  dlk_finalize<<<1, 288, 0, stream>>>(p, out);
}